// GCNLayer_4380866642245
// MI455X (gfx1250) — compile-verified
//
#include <hip/hip_runtime.h>
#include <hip/hip_bf16.h>

// Problem constants (match reference)
#define N_NODES 50000
#define N_EDGES 800000
#define D       128
#define D_OUT   128
#define K2      256            // 2*D concat width
#define XS_STRIDE 260          // 16x256 tile padded so ds_load_b64 per-lane rows hit distinct banks

typedef __attribute__((ext_vector_type(2))) float v2f;
typedef __attribute__((ext_vector_type(8))) float v8f;

// ---------------------------------------------------------------------------
// Kernel 1: zero the workspace (agg [N*D] + deg [N]) every launch
// ---------------------------------------------------------------------------
__global__ void gcn_zero_kernel(float* __restrict__ ws, int count) {
    int i = blockIdx.x * blockDim.x + threadIdx.x;
    if (i < count) ws[i] = 0.0f;
}

// ---------------------------------------------------------------------------
// Kernel 2: edge scatter. One wave32 per edge: 128-bit coalesced gather of
// feature[src], 4 global_atomic_add_f32 per lane into agg[dst], lane0 bumps deg.
// feature + agg are L2-resident (25.6 MB each << 192 MB L2).
// ---------------------------------------------------------------------------
__global__ void gcn_scatter_kernel(const float* __restrict__ feat,
                                   const int* __restrict__ src,
                                   const int* __restrict__ dst,
                                   float* __restrict__ agg,
                                   float* __restrict__ deg) {
    int wavesPerBlk = blockDim.x >> 5;
    int e = blockIdx.x * wavesPerBlk + (threadIdx.x >> 5);
    if (e >= N_EDGES) return;
    int lane = threadIdx.x & 31;
    int s = src[e];
    int t = dst[e];
    const float4 v = *(const float4*)(feat + (size_t)s * D + lane * 4);
    float* ap = agg + (size_t)t * D + lane * 4;
    atomicAdd(ap + 0, v.x);
    atomicAdd(ap + 1, v.y);
    atomicAdd(ap + 2, v.z);
    atomicAdd(ap + 3, v.w);
    if (lane == 0) atomicAdd(deg + t, 1.0f);
}

// ---------------------------------------------------------------------------
// Kernel 3: fused mean-normalize + concat + GEMM + bias using f32 WMMA.
// Block = 256 threads (8 waves) per 16-node row tile. LDS stages
// x[16][256] = [agg*invdeg | feature]; wave w computes output cols [16w,16w+16)
// with 64 chained v_wmma_f32_16x16x4_f32 (K = 256 in steps of 4).
// ---------------------------------------------------------------------------
__global__ void gcn_gemm_kernel(const float* __restrict__ feat,
                                const float* __restrict__ agg,
                                const float* __restrict__ deg,
                                const float* __restrict__ W,   // [D_OUT][K2] row-major
                                const float* __restrict__ bias,
                                float* __restrict__ out) {
    __shared__ float xs[16 * XS_STRIDE];
    __shared__ float invd[16];

    const int node0 = blockIdx.x * 16;
    const int t = threadIdx.x;

    if (t < 16) {
        float dg = deg[node0 + t];
        invd[t] = 1.0f / fmaxf(dg, 1.0f);
    }
    __syncthreads();

    // Stage x tile: 16 rows x 256 cols, 16 elements per thread
    #pragma unroll
    for (int i = t; i < 16 * K2; i += 256) {
        int r = i >> 8;
        int c = i & 255;
        float v;
        if (c < D) {
            v = agg[(size_t)(node0 + r) * D + c] * invd[r];
        } else {
            v = feat[(size_t)(node0 + r) * D + (c - D)];
        }
        xs[r * XS_STRIDE + c] = v;
    }
    __syncthreads();

    // Wave owns one 16x16 output tile (cols n0..n0+15)
    const int wave = t >> 5;          // 0..7
    const int lane = t & 31;
    const int half = lane >> 4;       // 0: K{0,1}, 1: K{2,3}
    const int lid  = lane & 15;       // M row (for A) / N col (for B)
    const int n0   = wave * 16;

    // B source: W^T pairs for this lane's output column (L2-resident W)
    const float* wrow = W + (size_t)(n0 + lid) * K2 + 2 * half;
    // A source: this lane's x row in LDS
    const float* arow = xs + lid * XS_STRIDE + 2 * half;

    __builtin_prefetch(wrow, 0, 3);   // global_prefetch_b8 into L2/WGP$

    v8f acc = {};
    #pragma unroll 8
    for (int k = 0; k < K2; k += 4) {
        v2f a = *(const v2f*)(arow + k);   // ds_load_b64
        v2f b = *(const v2f*)(wrow + k);   // global_load_b64
        // 8 args: (neg_a, A, neg_b, B, c_mod, C, reuse_a, reuse_b)
        acc = __builtin_amdgcn_wmma_f32_16x16x4_f32(
            false, a, false, b, (short)0, acc, false, false);
    }

    // C/D layout: VGPR j -> M = j + 8*half (row), N = lid (col)
    const float bv = bias[n0 + lid];
    #pragma unroll
    for (int j = 0; j < 8; ++j) {
        int row = node0 + j + 8 * half;
        out[(size_t)row * D_OUT + n0 + lid] = acc[j] + bv;
    }
}

// ---------------------------------------------------------------------------
extern "C" void kernel_launch(void* const* d_in, const int* in_sizes, int n_in,
                              void* d_out, int out_size, void* d_ws, size_t ws_size,
                              hipStream_t stream) {
    const float* feat = (const float*)d_in[0];
    const int*   src  = (const int*)d_in[1];
    const int*   dst  = (const int*)d_in[2];
    const float* W    = (const float*)d_in[3];
    const float* bias = (const float*)d_in[4];
    float*       out  = (float*)d_out;

    float* agg = (float*)d_ws;                       // N_NODES * D floats
    float* deg = agg + (size_t)N_NODES * D;          // N_NODES floats

    // 1) zero accumulators + degrees
    {
        int count = N_NODES * D + N_NODES;
        int blocks = (count + 255) / 256;
        gcn_zero_kernel<<<blocks, 256, 0, stream>>>(agg, count);
    }

    // 2) edge scatter: 8 waves (8 edges) per 256-thread block
    {
        int blocks = (N_EDGES + 7) / 8;
        gcn_scatter_kernel<<<blocks, 256, 0, stream>>>(feat, src, dst, agg, deg);
    }

    // 3) fused normalize + concat + WMMA GEMM + bias
    {
        int blocks = N_NODES / 16;   // 3125, exact tiling -> EXEC all-ones at WMMA
        gcn_gemm_kernel<<<blocks, 256, 0, stream>>>(feat, agg, deg, W, bias, out);
    }
}